// GNN_32229434589682
// MI455X (gfx1250) — compile-verified
//
#include <hip/hip_runtime.h>

#define NN 50000
#define EE 800000
#define DD 128
#define HD 64
#define GG 512
#define D_PAD 132   // 132 mod 64 = 4 -> conflict-free 16x4 fp32 A-fragment reads

typedef __attribute__((ext_vector_type(2))) float v2f;
typedef __attribute__((ext_vector_type(8))) float v8f;

// ---------------- graph preprocessing (CSR by dst) ----------------
__global__ __launch_bounds__(256) void k_zero_i(int* __restrict__ p, int n) {
    int i = blockIdx.x * 256 + threadIdx.x;
    if (i < n) p[i] = 0;
}

__global__ __launch_bounds__(256) void k_count(const long long* __restrict__ dstI,
                                               int* __restrict__ cnt_in) {
    int e = blockIdx.x * 256 + threadIdx.x;
    if (e < EE) atomicAdd(&cnt_in[(int)dstI[e]], 1);
}

__global__ __launch_bounds__(256) void k_dinv(const int* __restrict__ cnt_in,
                                              float* __restrict__ dinv) {
    int i = blockIdx.x * 256 + threadIdx.x;
    if (i < NN) dinv[i] = __frsqrt_rn((float)cnt_in[i] + 1.0f);  // +1 self loop
}

// single-block exclusive scan of cnt_in -> rowptr[NN+1]
__global__ __launch_bounds__(1024) void k_scan(const int* __restrict__ cnt_in,
                                               int* __restrict__ rowptr) {
    __shared__ int sh[1024];
    __shared__ int carry;
    if (threadIdx.x == 0) carry = 0;
    __syncthreads();
    for (int base = 0; base < NN; base += 1024) {
        int i = base + (int)threadIdx.x;
        int v = (i < NN) ? cnt_in[i] : 0;
        sh[threadIdx.x] = v;
        __syncthreads();
        for (int ofs = 1; ofs < 1024; ofs <<= 1) {
            int t = (threadIdx.x >= (unsigned)ofs) ? sh[threadIdx.x - ofs] : 0;
            __syncthreads();
            sh[threadIdx.x] += t;
            __syncthreads();
        }
        int incl = sh[threadIdx.x];
        if (i < NN) rowptr[i] = carry + incl - v;   // exclusive
        __syncthreads();
        if (threadIdx.x == 1023) carry += incl;     // block total
        __syncthreads();
    }
    if (threadIdx.x == 0) rowptr[NN] = carry;       // == EE
}

__global__ __launch_bounds__(256) void k_copy_i(const int* __restrict__ a,
                                                int* __restrict__ b, int n) {
    int i = blockIdx.x * 256 + threadIdx.x;
    if (i < n) b[i] = a[i];
}

__global__ __launch_bounds__(256) void k_fill(const long long* __restrict__ srcI,
                                              const long long* __restrict__ dstI,
                                              const float* __restrict__ dinv,
                                              int* __restrict__ cursor,
                                              int* __restrict__ col,
                                              float* __restrict__ wgt) {
    int e = blockIdx.x * 256 + threadIdx.x;
    if (e >= EE) return;
    int s = (int)srcI[e];
    int t = (int)dstI[e];
    int pos = atomicAdd(&cursor[t], 1);
    col[pos] = s;
    wgt[pos] = dinv[s] * dinv[t];
}

// ---------------- dense X@W via V_WMMA_F32_16X16X4_F32 ----------------
// grid.x = NN/16 = 3125 blocks, 256 threads = 8 waves; wave w owns cols [16w,16w+16)
__global__ __launch_bounds__(256) void k_matmul_wmma(const float* __restrict__ X,
                                                     const float* __restrict__ W,
                                                     float* __restrict__ Hout) {
    __shared__ float Alds[16 * D_PAD];
    const int row0 = blockIdx.x * 16;
    const int tid  = threadIdx.x;

    {   // cooperative coalesced load of the 16x128 A tile
        const float4* Xv = (const float4*)(X + (size_t)row0 * DD);
        #pragma unroll
        for (int i = tid; i < 512; i += 256) {
            int r = i >> 5;
            int c = i & 31;
            float4 v = Xv[(size_t)r * 32 + c];
            float* d = &Alds[r * D_PAD + c * 4];
            d[0] = v.x; d[1] = v.y; d[2] = v.z; d[3] = v.w;
        }
    }
    __syncthreads();

    const int wave = tid >> 5;
    const int lane = tid & 31;
    const int h    = lane >> 4;
    const int mn   = lane & 15;
    const int col0 = wave * 16;

    v8f c = {};
    #pragma unroll
    for (int k0 = 0; k0 < DD; k0 += 4) {
        const float* ap = &Alds[mn * D_PAD + k0 + 2 * h];
        v2f a; a[0] = ap[0]; a[1] = ap[1];
        v2f b;
        b[0] = W[(size_t)(k0 + 2 * h)     * DD + col0 + mn];
        b[1] = W[(size_t)(k0 + 2 * h + 1) * DD + col0 + mn];
        c = __builtin_amdgcn_wmma_f32_16x16x4_f32(false, a, false, b,
                                                  (short)0, c, false, false);
    }

    float* Hp = Hout + (size_t)row0 * DD + col0 + mn;
    #pragma unroll
    for (int r = 0; r < 8; ++r)
        Hp[(size_t)(r + 8 * h) * DD] = c[r];
}

// ------- fused neighbor gather + self loop + bias + relu: one wave per node -------
__global__ __launch_bounds__(256) void k_gather_bias_relu(const int* __restrict__ rowptr,
                                                          const int* __restrict__ col,
                                                          const float* __restrict__ wgt,
                                                          const float* __restrict__ dinv,
                                                          const float* __restrict__ hbuf,
                                                          const float* __restrict__ bias,
                                                          float* __restrict__ outv) {
    int t    = (blockIdx.x * 256 + threadIdx.x) >> 5;
    int lane = threadIdx.x & 31;
    if (t >= NN) return;

    float di = dinv[t];
    float sw = di * di;                               // self-loop norm
    float4 acc = ((const float4*)(hbuf + (size_t)t * DD))[lane];
    acc.x *= sw; acc.y *= sw; acc.z *= sw; acc.w *= sw;

    int k0 = rowptr[t], k1 = rowptr[t + 1];
    for (int k = k0; k < k1; ++k) {
        int   s = col[k];
        float w = wgt[k];
        float4 hv = ((const float4*)(hbuf + (size_t)s * DD))[lane];
        acc.x += hv.x * w; acc.y += hv.y * w;
        acc.z += hv.z * w; acc.w += hv.w * w;
    }

    float4 bv = ((const float4*)bias)[lane];
    acc.x = fmaxf(acc.x + bv.x, 0.0f);
    acc.y = fmaxf(acc.y + bv.y, 0.0f);
    acc.z = fmaxf(acc.z + bv.z, 0.0f);
    acc.w = fmaxf(acc.w + bv.w, 0.0f);
    ((float4*)(outv + (size_t)t * DD))[lane] = acc;
}

// ---------------- pooling: batch is sorted -> binary-search boundaries ----------------
__global__ __launch_bounds__(256) void k_bounds(const long long* __restrict__ batch,
                                                int* __restrict__ gstart) {
    int g = blockIdx.x * 256 + threadIdx.x;
    if (g > GG) return;
    int lo = 0, hi = NN;                              // first i with batch[i] >= g
    while (lo < hi) {
        int mid = (lo + hi) >> 1;
        if ((int)batch[mid] < g) lo = mid + 1; else hi = mid;
    }
    gstart[g] = lo;
}

__global__ __launch_bounds__(128) void k_pool_mean(const float* __restrict__ cur,
                                                   const int* __restrict__ gstart,
                                                   float* __restrict__ pooledmean) {
    int g = blockIdx.x, d = threadIdx.x;
    int s0 = gstart[g], s1 = gstart[g + 1];
    float acc = 0.0f;
    for (int i = s0; i < s1; ++i) acc += cur[(size_t)i * DD + d];
    float c = fmaxf((float)(s1 - s0), 1.0f);
    pooledmean[g * DD + d] = acc / c;
}

// ---------------- classifier head: relu(mean @ Wc + bc) @ Wo + bo ----------------
__global__ __launch_bounds__(64) void k_classifier(const float* __restrict__ pooledmean,
                                                   const float* __restrict__ Wc,
                                                   const float* __restrict__ bc,
                                                   const float* __restrict__ Wo,
                                                   const float* __restrict__ bo,
                                                   float* __restrict__ outv) {
    __shared__ float mean[DD];
    __shared__ float zred[HD];
    int g = blockIdx.x, j = threadIdx.x;
    for (int k = j; k < DD; k += HD) mean[k] = pooledmean[g * DD + k];
    __syncthreads();
    float acc = bc[j];
    #pragma unroll 8
    for (int k = 0; k < DD; ++k) acc += mean[k] * Wc[k * HD + j];
    zred[j] = fmaxf(acc, 0.0f) * Wo[j];
    __syncthreads();
    if (j == 0) {
        float s = 0.0f;
        #pragma unroll
        for (int t = 0; t < HD; ++t) s += zred[t];
        outv[g] = s + bo[0];
    }
}

static inline size_t al256(size_t x) { return (x + 255) & ~size_t(255); }

extern "C" void kernel_launch(void* const* d_in, const int* in_sizes, int n_in,
                              void* d_out, int out_size, void* d_ws, size_t ws_size,
                              hipStream_t stream) {
    const float*     x     = (const float*)d_in[0];
    const long long* ei    = (const long long*)d_in[1];   // (2,E) int64
    const long long* batch = (const long long*)d_in[2];
    const float* W1 = (const float*)d_in[3];
    const float* b1 = (const float*)d_in[4];
    const float* W2 = (const float*)d_in[5];
    const float* b2 = (const float*)d_in[6];
    const float* W3 = (const float*)d_in[7];
    const float* b3 = (const float*)d_in[8];
    const float* Wc = (const float*)d_in[9];
    const float* bc = (const float*)d_in[10];
    const float* Wo = (const float*)d_in[11];
    const float* bo = (const float*)d_in[12];
    float* outp = (float*)d_out;

    const long long* srcI = ei;        // edge_index[0]
    const long long* dstI = ei + EE;   // edge_index[1]

    char* ws = (char*)d_ws;
    size_t off = 0;
    int*   cnt_in  = (int*)  (ws + off); off += al256((size_t)NN * 4);
    int*   rowptr  = (int*)  (ws + off); off += al256((size_t)(NN + 1) * 4);
    int*   cursor  = (int*)  (ws + off); off += al256((size_t)NN * 4);
    int*   col     = (int*)  (ws + off); off += al256((size_t)EE * 4);
    float* wgt     = (float*)(ws + off); off += al256((size_t)EE * 4);
    float* dinv    = (float*)(ws + off); off += al256((size_t)NN * 4);
    float* hbuf    = (float*)(ws + off); off += al256((size_t)NN * DD * 4);
    float* curbuf  = (float*)(ws + off); off += al256((size_t)NN * DD * 4);
    int*   gstart  = (int*)  (ws + off); off += al256((size_t)(GG + 1) * 4);
    float* pooledm = (float*)(ws + off); off += al256((size_t)GG * DD * 4);
    (void)ws_size; (void)in_sizes; (void)n_in; (void)out_size;

    // ---- build normalization + CSR (once; reused by all 3 layers) ----
    k_zero_i<<<(NN + 255) / 256, 256, 0, stream>>>(cnt_in, NN);
    k_count <<<(EE + 255) / 256, 256, 0, stream>>>(dstI, cnt_in);
    k_dinv  <<<(NN + 255) / 256, 256, 0, stream>>>(cnt_in, dinv);
    k_scan  <<<1, 1024, 0, stream>>>(cnt_in, rowptr);
    k_copy_i<<<(NN + 255) / 256, 256, 0, stream>>>(rowptr, cursor, NN);
    k_fill  <<<(EE + 255) / 256, 256, 0, stream>>>(srcI, dstI, dinv, cursor, col, wgt);

    // ---- three GCN layers ----
    const float* Wl[3] = {W1, W2, W3};
    const float* bl[3] = {b1, b2, b3};
    const float* xcur = x;
    const int GATHER_BLOCKS = (NN * 32 + 255) / 256;   // one wave per node
    for (int l = 0; l < 3; ++l) {
        k_matmul_wmma<<<NN / 16, 256, 0, stream>>>(xcur, Wl[l], hbuf);
        k_gather_bias_relu<<<GATHER_BLOCKS, 256, 0, stream>>>(rowptr, col, wgt, dinv,
                                                              hbuf, bl[l], curbuf);
        xcur = curbuf;
    }

    // ---- mean pool + head ----
    k_bounds    <<<(GG + 1 + 255) / 256, 256, 0, stream>>>(batch, gstart);
    k_pool_mean <<<GG, 128, 0, stream>>>(curbuf, gstart, pooledm);
    k_classifier<<<GG, 64, 0, stream>>>(pooledm, Wc, bc, Wo, bo, outp);
}